// TripletSoftmaxLoss_17102559773290
// MI455X (gfx1250) — compile-verified
//
#include <hip/hip_runtime.h>
#include <math.h>

#define EPS     1e-6f
#define LAMBDA  0.1f
#define BATCH   8192
#define DIM     512
#define NCLASS  1000

typedef float v2f __attribute__((ext_vector_type(2)));
typedef float v8f __attribute__((ext_vector_type(8)));

// ---------------------------------------------------------------------------
// Triplet term: per-row ||a-p+eps||^2 and ||a-n+eps||^2 via WMMA reduction.
// One wave per 16-row tile. A = 16x4 tile of squared diffs, B = ones(4x16),
// so C[m,n] accumulates the row sum (replicated across n).
// A-matrix f32 16x4 layout (ISA 7.12.2): lanes 0-15 -> K=k+{0,1}, lanes
// 16-31 -> K=k+{2,3}, M = lane & 15.
// C layout: VGPR i, lanes 0-15 -> M=i; lanes 16-31 -> M=8+i.
// ---------------------------------------------------------------------------
__global__ __launch_bounds__(256) void triplet_kernel(
    const float* __restrict__ anchor,
    const float* __restrict__ positive,
    const float* __restrict__ negative,
    float* __restrict__ acc /* acc[0] = triplet sum */)
{
  const int wave = (blockIdx.x * blockDim.x + threadIdx.x) >> 5;
  const int lane = threadIdx.x & 31;
  const int row  = wave * 16 + (lane & 15);          // 512 waves cover 8192 rows
  const int koff = (lane >> 4) * 2;                  // 0 for lanes 0-15, 2 for 16-31
  const int base = row * DIM + koff;

  v2f ones2; ones2.x = 1.0f; ones2.y = 1.0f;
  v8f c_ap = {}; v8f c_an = {};

  for (int k = 0; k < DIM; k += 4) {
    float a0 = anchor[base + k];
    float a1 = anchor[base + k + 1];
    float p0 = positive[base + k];
    float p1 = positive[base + k + 1];
    float n0 = negative[base + k];
    float n1 = negative[base + k + 1];
    float dp0 = a0 - p0 + EPS, dp1 = a1 - p1 + EPS;
    float dn0 = a0 - n0 + EPS, dn1 = a1 - n1 + EPS;
    v2f ap; ap.x = dp0 * dp0; ap.y = dp1 * dp1;
    v2f an; an.x = dn0 * dn0; an.y = dn1 * dn1;
    // 8 args: (neg_a, A, neg_b, B, c_mod, C, reuse_a, reuse_b)
    c_ap = __builtin_amdgcn_wmma_f32_16x16x4_f32(false, ap, false, ones2,
                                                 (short)0, c_ap, false, false);
    c_an = __builtin_amdgcn_wmma_f32_16x16x4_f32(false, an, false, ones2,
                                                 (short)0, c_an, false, false);
  }

  // Lane 0 holds rows 0..7 sums in c[0..7]; lane 16 holds rows 8..15.
  float part = 0.0f;
#pragma unroll
  for (int i = 0; i < 8; ++i) {
    float dap = sqrtf(c_ap[i]);
    float dan = sqrtf(c_an[i]);
    part += fmaxf(dap - dan, 0.0f);
  }
  if (lane == 0 || lane == 16)
    atomicAdd(acc, part);
}

// ---------------------------------------------------------------------------
// Cross entropy (mean): one wave32 per row, single-pass online softmax.
// ---------------------------------------------------------------------------
__global__ __launch_bounds__(256) void ce_kernel(
    const float* __restrict__ outputs,
    const int*   __restrict__ labels,
    float* __restrict__ acc /* acc[1] = sum of per-row CE */)
{
  const int row  = blockIdx.x * (blockDim.x >> 5) + (threadIdx.x >> 5);
  const int lane = threadIdx.x & 31;
  if (row >= BATCH) return;
  const float* __restrict__ x = outputs + (size_t)row * NCLASS;

  float m = -INFINITY;
  float s = 0.0f;
  for (int c = lane; c < NCLASS; c += 32) {
    float v  = x[c];
    float mn = fmaxf(m, v);
    s = s * __expf(m - mn) + __expf(v - mn);
    m = mn;
  }
  // wave32 combine of (max, scaled-sum) pairs
  for (int off = 16; off > 0; off >>= 1) {
    float m2 = __shfl_xor(m, off, 32);
    float s2 = __shfl_xor(s, off, 32);
    float mn = fmaxf(m, m2);
    s = s * __expf(m - mn) + s2 * __expf(m2 - mn);
    m = mn;
  }
  if (lane == 0) {
    float logit = x[labels[row]];          // hits L2 (already streamed)
    float loss  = (m + __logf(s)) - logit; // logsumexp - label logit
    atomicAdd(acc + 1, loss);
  }
}

// ---------------------------------------------------------------------------
__global__ void init_kernel(float* acc) {
  acc[0] = 0.0f;
  acc[1] = 0.0f;
}

__global__ void finalize_kernel(const float* __restrict__ acc,
                                float* __restrict__ out) {
  float trip = acc[0];
  float ce   = acc[1] * (1.0f / (float)BATCH);
  out[0] = ce + LAMBDA * trip;  // loss_total
  out[1] = trip;                // loss_triplet
  out[2] = ce;                  // loss_softmax
}

// ---------------------------------------------------------------------------
extern "C" void kernel_launch(void* const* d_in, const int* in_sizes, int n_in,
                              void* d_out, int out_size, void* d_ws, size_t ws_size,
                              hipStream_t stream) {
  const float* anchor   = (const float*)d_in[0];
  const float* positive = (const float*)d_in[1];
  const float* negative = (const float*)d_in[2];
  const float* outputs  = (const float*)d_in[3];
  const int*   labels   = (const int*)d_in[4];
  float* out = (float*)d_out;
  float* acc = (float*)d_ws;   // acc[0]=triplet sum, acc[1]=ce sum

  init_kernel<<<1, 1, 0, stream>>>(acc);

  // 8192 rows / 16 rows-per-wave / 8 waves-per-block = 64 blocks
  triplet_kernel<<<64, 256, 0, stream>>>(anchor, positive, negative, acc);

  // 8 rows per 256-thread block -> 1024 blocks
  ce_kernel<<<BATCH / 8, 256, 0, stream>>>(outputs, labels, acc);

  finalize_kernel<<<1, 1, 0, stream>>>(acc, out);
}